// TransformerConvLayer_38156489458109
// MI455X (gfx1250) — compile-verified
//
#include <hip/hip_runtime.h>
#include <hip/hip_bf16.h>
#include <stdint.h>

// ---------------------------------------------------------------------------
// TransformerConv (PyG, concat=True, root_weight=True) for MI455X / gfx1250.
//   N=100000 nodes, C=128, H=8 heads, D=16.  4 GEMMs via V_WMMA_F32_16X16X4_F32,
//   edge phase = gather (L2-resident: q,k,v = 153.6MB < 192MB L2) + segment
//   softmax + atomic scatter.
// ---------------------------------------------------------------------------

typedef __attribute__((ext_vector_type(2))) float v2f;
typedef __attribute__((ext_vector_type(8))) float v8f;

#define CCH 128
#define HH  8
#define HD  16

// ---- monotonic float<->uint encoding for atomic segment-max ----------------
__device__ __forceinline__ unsigned fenc(float f) {
    unsigned u = __float_as_uint(f);
    return (u & 0x80000000u) ? ~u : (u | 0x80000000u);
}
__device__ __forceinline__ float fdec(unsigned u) {
    return (u & 0x80000000u) ? __uint_as_float(u & 0x7fffffffu)
                             : __uint_as_float(~u);
}
#define ENC_NEG_INF 0x007FFFFFu  // fenc(-inf)

// ---------------------------------------------------------------------------
// Fused QKVS GEMM:  out = x @ W + b  for 4 weight sets (blockIdx.y selects).
// Block = 256 threads = 8 waves; each wave computes one 16x16 tile of the
// 16-row x 128-col block. K-loop: 32 x v_wmma_f32_16x16x4_f32.
// ---------------------------------------------------------------------------
__global__ void __launch_bounds__(256)
tc_gemm_qkvs(const float* __restrict__ x,
             const float* __restrict__ Wq, const float* __restrict__ bq,
             const float* __restrict__ Wk, const float* __restrict__ bk,
             const float* __restrict__ Wv, const float* __restrict__ bv,
             const float* __restrict__ Ws, const float* __restrict__ bs,
             float* __restrict__ q, float* __restrict__ k,
             float* __restrict__ v, float* __restrict__ s,
             int N)
{
    const float* W;  const float* bias;  float* out;
    switch (blockIdx.y) {
        case 0:  W = Wq; bias = bq; out = q; break;
        case 1:  W = Wk; bias = bk; out = k; break;
        case 2:  W = Wv; bias = bv; out = v; break;
        default: W = Ws; bias = bs; out = s; break;
    }

    const int wave = threadIdx.x >> 5;          // 0..7 -> 16-col tile
    const int lane = threadIdx.x & 31;
    const int row0 = blockIdx.x * 16;

    const int m   = lane & 15;                  // A row within tile
    const int kk  = (lane >> 4) * 2;            // K sub-offset {0,2}
    const int col = wave * 16 + (lane & 15);    // B/D column

    int arow = row0 + m;                        // clamp OOB loads (keep EXEC=all-1)
    if (arow >= N) arow = N - 1;
    const float* xrow = x + (size_t)arow * CCH;

    v8f acc = {};
    #pragma unroll
    for (int kb = 0; kb < CCH; kb += 4) {
        v2f a = *(const v2f*)(xrow + kb + kk);                 // A[m][kb+kk .. +1]
        v2f b;
        b.x = W[(size_t)(kb + kk)     * CCH + col];            // B[kb+kk  ][col]
        b.y = W[(size_t)(kb + kk + 1) * CCH + col];            // B[kb+kk+1][col]
        acc = __builtin_amdgcn_wmma_f32_16x16x4_f32(
                  false, a, false, b, (short)0, acc, false, false);
    }

    const float bv_ = bias[col];
    #pragma unroll
    for (int r = 0; r < 8; ++r) {
        const int mrow = r + 8 * (lane >> 4);   // C/D: VGPR r -> rows r / r+8
        const int grow = row0 + mrow;
        if (grow < N)
            out[(size_t)grow * CCH + col] = acc[r] + bv_;
    }
}

// ---------------------------------------------------------------------------
// Init per-(node,head) stats.
// ---------------------------------------------------------------------------
__global__ void tc_init_stats(unsigned* __restrict__ amaxU,
                              float* __restrict__ denom, int n)
{
    int i = blockIdx.x * blockDim.x + threadIdx.x;
    if (i < n) { amaxU[i] = ENC_NEG_INF; denom[i] = 0.0f; }
}

// ---------------------------------------------------------------------------
// Per-edge attention logits + segment max.  One wave per edge:
// lane -> head (lane>>2), 4 lanes x float4 cover the 16-dim head.
// ---------------------------------------------------------------------------
__global__ void __launch_bounds__(256)
tc_edge_alpha(const int* __restrict__ ei,
              const float* __restrict__ q, const float* __restrict__ k,
              float* __restrict__ alphaBuf, unsigned* __restrict__ amaxU,
              int E)
{
    const int e = blockIdx.x * 8 + (threadIdx.x >> 5);
    if (e >= E) return;
    const int lane = threadIdx.x & 31;
    const int src = ei[e];
    const int dst = ei[(size_t)E + e];
    const int h  = lane >> 2;
    const int dp = (lane & 3) * 4;

    const float4 qv = *(const float4*)(q + (size_t)dst * CCH + h * HD + dp);
    const float4 kv = *(const float4*)(k + (size_t)src * CCH + h * HD + dp);
    float p = qv.x * kv.x + qv.y * kv.y + qv.z * kv.z + qv.w * kv.w;
    p += __shfl_xor(p, 1, 32);
    p += __shfl_xor(p, 2, 32);

    if ((lane & 3) == 0) {
        const float a = p * 0.25f;               // 1/sqrt(D), D=16
        alphaBuf[(size_t)e * HH + h] = a;
        atomicMax(&amaxU[(size_t)dst * HH + h], fenc(a));
    }
}

// ---------------------------------------------------------------------------
// exp(alpha - max) and segment denominator.  Thread per (edge, head).
// ---------------------------------------------------------------------------
__global__ void __launch_bounds__(256)
tc_edge_expsum(const int* __restrict__ ei,
               float* __restrict__ alphaBuf,
               const unsigned* __restrict__ amaxU,
               float* __restrict__ denom, int E)
{
    const int i = blockIdx.x * blockDim.x + threadIdx.x;
    if (i >= E * HH) return;
    const int e = i >> 3, h = i & 7;
    const int dst = ei[(size_t)E + e];
    const float mx = fdec(amaxU[(size_t)dst * HH + h]);
    const float ea = __expf(alphaBuf[i] - mx);
    alphaBuf[i] = ea;                            // reuse buffer in-place
    atomicAdd(&denom[(size_t)dst * HH + h], ea);
}

// ---------------------------------------------------------------------------
// Weighted scatter of values into out (which already holds x@Ws + bs).
// One wave per edge; lane covers 4 channels (float4), head = lane>>2.
// ---------------------------------------------------------------------------
__global__ void __launch_bounds__(256)
tc_edge_scatter(const int* __restrict__ ei,
                const float* __restrict__ v,
                const float* __restrict__ alphaBuf,
                const float* __restrict__ denom,
                float* __restrict__ out, int E)
{
    const int e = blockIdx.x * 8 + (threadIdx.x >> 5);
    if (e >= E) return;
    const int lane = threadIdx.x & 31;
    const int src = ei[e];
    const int dst = ei[(size_t)E + e];
    const int h = lane >> 2;

    const float w = alphaBuf[(size_t)e * HH + h] /
                    (denom[(size_t)dst * HH + h] + 1e-16f);

    const float4 vv = *(const float4*)(v + (size_t)src * CCH + lane * 4);
    float* op = out + (size_t)dst * CCH + lane * 4;
    atomicAdd(op + 0, w * vv.x);
    atomicAdd(op + 1, w * vv.y);
    atomicAdd(op + 2, w * vv.z);
    atomicAdd(op + 3, w * vv.w);
}

// ---------------------------------------------------------------------------
extern "C" void kernel_launch(void* const* d_in, const int* in_sizes, int n_in,
                              void* d_out, int out_size, void* d_ws, size_t ws_size,
                              hipStream_t stream)
{
    const float* x  = (const float*)d_in[0];
    const int*   ei = (const int*)  d_in[1];
    const float* Wq = (const float*)d_in[2]; const float* bq = (const float*)d_in[3];
    const float* Wk = (const float*)d_in[4]; const float* bk = (const float*)d_in[5];
    const float* Wv = (const float*)d_in[6]; const float* bv = (const float*)d_in[7];
    const float* Ws = (const float*)d_in[8]; const float* bs = (const float*)d_in[9];
    float* out = (float*)d_out;

    const int N = in_sizes[0] / CCH;      // 100000
    const int E = in_sizes[1] / 2;        // 1600000

    // workspace carve-up
    char* ws = (char*)d_ws;
    const size_t nodeBytes = (size_t)N * CCH * sizeof(float);
    float*    q        = (float*)ws;                 ws += nodeBytes;
    float*    k        = (float*)ws;                 ws += nodeBytes;
    float*    v        = (float*)ws;                 ws += nodeBytes;
    float*    alphaBuf = (float*)ws;                 ws += (size_t)E * HH * sizeof(float);
    unsigned* amaxU    = (unsigned*)ws;              ws += (size_t)N * HH * sizeof(unsigned);
    float*    denom    = (float*)ws;

    const int rowTiles = (N + 15) / 16;

    tc_gemm_qkvs<<<dim3(rowTiles, 4), 256, 0, stream>>>(
        x, Wq, bq, Wk, bk, Wv, bv, Ws, bs, q, k, v, out, N);

    tc_init_stats<<<(N * HH + 255) / 256, 256, 0, stream>>>(amaxU, denom, N * HH);

    tc_edge_alpha<<<(E + 7) / 8, 256, 0, stream>>>(ei, q, k, alphaBuf, amaxU, E);

    tc_edge_expsum<<<(E * HH + 255) / 256, 256, 0, stream>>>(ei, alphaBuf, amaxU, denom, E);

    tc_edge_scatter<<<(E + 7) / 8, 256, 0, stream>>>(ei, v, alphaBuf, denom, out, E);
}